// ACArLModule_74311524155916
// MI455X (gfx1250) — compile-verified
//
#include <hip/hip_runtime.h>
#include <math.h>

// ---------------------------------------------------------------------------
// Problem constants (match reference)
// ---------------------------------------------------------------------------
#define B_     8
#define D_     768
#define HH     32
#define WW     32
#define HW_    1024
#define C_     2
#define K_     8
#define CD     128            // COORD_DIM
#define IN_DIM 1664           // 2*D + CD
#define H1DIM  256
#define H2DIM  128
#define NPTS   256            // 16x16 sampled grid points
#define NOFF   12             // neighbor offsets with |di|+|dj| in {2,4}
#define SLOTS  3072           // NPTS*NOFF padded pair slots (2756 valid)
#define BCN    (B_*C_)        // 16

typedef float v2f __attribute__((ext_vector_type(2)));
typedef float v8f __attribute__((ext_vector_type(8)));

__device__ const int OFFT[NOFF][2] = {
  {0,2},{0,-2},{2,0},{-2,0},
  {0,4},{0,-4},{4,0},{-4,0},
  {2,2},{2,-2},{-2,2},{-2,-2}
};

// ---------------------------------------------------------------------------
// Pair table: slot = q*12 + s, q = grid point (16x16, stride 2)
// ---------------------------------------------------------------------------
__global__ void init_pairs(int* __restrict__ ia, int* __restrict__ ib,
                           int* __restrict__ valid) {
  int t = blockIdx.x * blockDim.x + threadIdx.x;
  if (t >= SLOTS) return;
  int q = t / NOFF, s = t % NOFF;
  int pi = (q >> 4) * 2, pj = (q & 15) * 2;
  int ni = pi + OFFT[s][0], nj = pj + OFFT[s][1];
  int ok = (ni >= 0) && (ni < HH) && (nj >= 0) && (nj < WW);
  valid[t] = ok;
  ia[t] = pi * WW + pj;
  ib[t] = ok ? (ni * WW + nj) : 0;
}

// ---------------------------------------------------------------------------
// Fused gather for x[p,k] = [features(idx_a) | features(idx_b) | coord_emb avg]
// ---------------------------------------------------------------------------
__device__ __forceinline__ float x_gather(const float* __restrict__ featb,
                                          const float* __restrict__ coord_emb,
                                          const int* __restrict__ ia,
                                          const int* __restrict__ ib,
                                          const int* __restrict__ valid,
                                          int p, int k) {
  if (!valid[p]) return 0.f;
  int a = ia[p], b = ib[p];
  if (k < D_)       return featb[(size_t)k * HW_ + a];
  if (k < 2 * D_)   return featb[(size_t)(k - D_) * HW_ + b];
  int e = k - 2 * D_;
  return 0.5f * (coord_emb[(size_t)a * CD + e] + coord_emb[(size_t)b * CD + e]);
}

// ---------------------------------------------------------------------------
// fp32 WMMA GEMM:  Out[bc, m, n] = relu( sum_k A[m,k] * W[c,k,n] + bias[c,n] )
//   MODE 0: A comes from fused gather (GEMM1, K=1664, N=256)
//   MODE 1: A = h1[bc]                (GEMM2, K=256,  N=128)
// 128x64 block tile, 8 waves: wave w owns one 16-row stripe x four 16-col
// tiles (full BN). K-pairs are stored contiguously in LDS so every WMMA
// operand fragment is one aligned ds_load_b64 into an even VGPR pair.
// ---------------------------------------------------------------------------
#define BM 128
#define BN 64
#define KC 16

template <int MODE>
__global__ __launch_bounds__(256)
void wmma_gemm(const float* __restrict__ Ain,   // MODE 1: (BC, M, K)
               const float* __restrict__ Wt,    // (C, K, N)
               const float* __restrict__ bias,  // (C, N)
               float* __restrict__ Out,         // (BC, M, N)
               const float* __restrict__ features,
               const float* __restrict__ coord_emb,
               const int* __restrict__ ia, const int* __restrict__ ib,
               const int* __restrict__ valid,
               int Mdim, int Ndim, int Kdim) {
  const int bc = blockIdx.z;
  const int bidx = bc >> 1;
  const int c = bc & 1;
  const int m0 = blockIdx.x * BM;
  const int n0 = blockIdx.y * BN;

  // As[m][k]: row stride 18 floats (72 B) -> every even-k pair is 8B aligned
  //           and the 16 same-column readers land on distinct banks.
  // Bs[kp][2n+h] = { W[2kp][n], W[2kp+1][n] }: one b64 load per B fragment.
  __shared__ float As[BM][KC + 2];
  __shared__ float Bs[KC / 2][BN * 2 + 2];

  const int tid  = threadIdx.x;
  const int wave = tid >> 5;          // 0..7 (wave32) -> row stripe
  const int lane = tid & 31;

  const int lrow  = lane & 15;        // M (for A/D) or N (for B/D) within tile
  const int khalf = (lane >> 4) * 2;  // K base within group of 4 for this half

  v8f acc[4] = {};

  const float* featb = features + (size_t)bidx * D_ * HW_;
  const float* Wc    = Wt + (size_t)c * Kdim * Ndim;

  for (int kk = 0; kk < Kdim; kk += KC) {
    // Stage A tile (BM x KC): m-major so neighboring threads gather nearby rows.
    for (int t = tid; t < BM * KC; t += 256) {
      int m = t & (BM - 1), kc = t >> 7;
      float v;
      if (MODE == 0)
        v = x_gather(featb, coord_emb, ia, ib, valid, m0 + m, kk + kc);
      else
        v = Ain[((size_t)bc * Mdim + m0 + m) * Kdim + kk + kc];
      As[m][kc] = v;
    }
    // Stage B tile (KC x BN) in k-paired layout; prefetch next chunk rows.
    const bool more = (kk + KC) < Kdim;
    for (int t = tid; t < KC * BN; t += 256) {
      int n = t & 63, kc = t >> 6;
      Bs[kc >> 1][n * 2 + (kc & 1)] = Wc[(size_t)(kk + kc) * Ndim + n0 + n];
      if (more)
        __builtin_prefetch(&Wc[(size_t)(kk + KC + kc) * Ndim + n0 + n], 0, 1);
    }
    if (MODE == 1 && more) {
      // Prefetch next A chunk too (contiguous rows of h1).
      for (int t = tid; t < BM * KC; t += 256) {
        int m = t & (BM - 1), kc = t >> 7;
        __builtin_prefetch(&Ain[((size_t)bc * Mdim + m0 + m) * Kdim + kk + KC + kc], 0, 1);
      }
    }
    __syncthreads();

#pragma unroll
    for (int ks = 0; ks < KC / 4; ++ks) {
      // A fragment: K = 4*ks + khalf + {0,1} -- one aligned b64 LDS load.
      v2f a = *(const v2f*)&As[wave * 16 + lrow][ks * 4 + khalf];
#pragma unroll
      for (int tt = 0; tt < 4; ++tt) {
        // B fragment: same K pair, N = tt*16 + lrow -- one aligned b64 load.
        v2f b = *(const v2f*)&Bs[ks * 2 + (lane >> 4)][(tt * 16 + lrow) * 2];
        acc[tt] = __builtin_amdgcn_wmma_f32_16x16x4_f32(false, a, false, b,
                                                        (short)0, acc[tt],
                                                        false, false);
      }
    }
    __syncthreads();
  }

  // Epilogue: D layout — VGPR r: lanes 0-15 M=r, lanes 16-31 M=r+8; N=lane&15.
  const int rowhalf = (lane >> 4) * 8;
#pragma unroll
  for (int tt = 0; tt < 4; ++tt) {
    const int gn = n0 + tt * 16 + lrow;
    const float bv = bias[c * Ndim + gn];
#pragma unroll
    for (int r = 0; r < 8; ++r) {
      int gm = m0 + wave * 16 + r + rowhalf;
      Out[((size_t)bc * Mdim + gm) * Ndim + gn] = fmaxf(acc[tt][r] + bv, 0.f);
    }
  }
}

// ---------------------------------------------------------------------------
// GEMM3 + sigmoid: val[bc,p] = sigmoid( h2[bc,p,:] . W3[c,:] + b3[c] )
// One wave per row, shuffle reduction.
// ---------------------------------------------------------------------------
__global__ __launch_bounds__(256)
void gemm3_val(const float* __restrict__ h2, const float* __restrict__ W3,
               const float* __restrict__ b3, float* __restrict__ val) {
  int wave = threadIdx.x >> 5, lane = threadIdx.x & 31;
  int row = blockIdx.x * 8 + wave;          // [0, BCN*SLOTS)
  int c = (row / SLOTS) & 1;
  const float* h = h2 + (size_t)row * H2DIM;
  const float* wv = W3 + c * H2DIM;
  float s = 0.f;
  for (int i = lane; i < H2DIM; i += 32) s += h[i] * wv[i];
  for (int off = 16; off; off >>= 1) s += __shfl_down(s, off, 32);
  if (lane == 0) val[row] = 1.f / (1.f + expf(-(s + b3[c])));
}

// rowsum[bc,q] = 1 + sum of valid pair values emitted by point q
__global__ void row_sums(const float* __restrict__ val, const int* __restrict__ valid,
                         float* __restrict__ rowsum) {
  int bc = blockIdx.x, q = threadIdx.x;
  float s = 1.f;
#pragma unroll
  for (int k = 0; k < NOFF; ++k) {
    int slot = q * NOFF + k;
    if (valid[slot]) s += val[(size_t)bc * SLOTS + slot];
  }
  rowsum[bc * NPTS + q] = s;
}

// Dense A: identity rows scaled by 1/(rowsum+1e-8); float4 stores (67MB -> ~3us)
__global__ void fill_A(const float* __restrict__ rowsum, float* __restrict__ A) {
  size_t t = (size_t)blockIdx.x * blockDim.x + threadIdx.x;   // BCN*1024*256
  if (t >= (size_t)BCN * HW_ * (HW_ / 4)) return;
  int j4 = (int)(t & 255) * 4;
  int i  = (int)((t >> 8) & 1023);
  int bc = (int)(t >> 18);
  float diag;
  int pi = i >> 5, pj = i & 31;
  if (!(pi & 1) && !(pj & 1))
    diag = 1.f / (rowsum[bc * NPTS + (pi >> 1) * 16 + (pj >> 1)] + 1e-8f);
  else
    diag = 1.f / (1.f + 1e-8f);
  float4 v = make_float4(0.f, 0.f, 0.f, 0.f);
  if (i >= j4 && i < j4 + 4) (&v.x)[i - j4] = diag;
  reinterpret_cast<float4*>(A)[t] = v;
}

__global__ void scatter_A(const float* __restrict__ val, const float* __restrict__ rowsum,
                          const int* __restrict__ ia, const int* __restrict__ ib,
                          const int* __restrict__ valid, float* __restrict__ A) {
  int t = blockIdx.x * blockDim.x + threadIdx.x;   // BCN*SLOTS
  if (t >= BCN * SLOTS) return;
  int bc = t / SLOTS, slot = t % SLOTS;
  if (!valid[slot]) return;
  int q = slot / NOFF;
  float inv = 1.f / (rowsum[bc * NPTS + q] + 1e-8f);
  A[(((size_t)bc * HW_) + ia[slot]) * HW_ + ib[slot]] = val[t] * inv;
}

// ---------------------------------------------------------------------------
// Diffusion on the closed 256-point subsystem (off-grid rows have L == 0).
// ---------------------------------------------------------------------------
__global__ __launch_bounds__(256)
void diffusion(const float* __restrict__ cam_raw, const float* __restrict__ val,
               const float* __restrict__ rowsum, const int* __restrict__ valid,
               float* __restrict__ msamp) {
  int bc = blockIdx.x, q = threadIdx.x;
  __shared__ float Ms[2][NPTS];
  int pi = (q >> 4) * 2, pj = (q & 15) * 2;
  float rs  = rowsum[bc * NPTS + q];
  float inv = 1.f / (rs + 1e-8f);
  float diag = inv, deg = rs * inv;
  int   nbq[NOFF]; float av[NOFF];
#pragma unroll
  for (int s = 0; s < NOFF; ++s) {
    int slot = q * NOFF + s;
    if (valid[slot]) {
      int ni = pi + OFFT[s][0], nj = pj + OFFT[s][1];
      nbq[s] = (ni >> 1) * 16 + (nj >> 1);
      av[s]  = val[(size_t)bc * SLOTS + slot] * inv;
    } else { nbq[s] = 0; av[s] = 0.f; }
  }
  Ms[0][q] = cam_raw[bc * HW_ + pi * WW + pj];
  __syncthreads();
  int cur = 0;
  for (int step = 0; step < 3; ++step) {
    float m = Ms[cur][q];
    float am = diag * m;
#pragma unroll
    for (int s = 0; s < NOFF; ++s) am += av[s] * Ms[cur][nbq[s]];
    Ms[cur ^ 1][q] = m + 0.1f * (am - deg * m);
    cur ^= 1;
    __syncthreads();
  }
  msamp[bc * NPTS + q] = Ms[cur][q];
}

__global__ void finalize_cam(const float* __restrict__ cam_raw, const float* __restrict__ msamp,
                             const float* __restrict__ scale, const float* __restrict__ biasp,
                             float* __restrict__ out_cam, float* __restrict__ out_conf) {
  int t = blockIdx.x * blockDim.x + threadIdx.x;   // BCN*HW
  if (t >= BCN * HW_) return;
  int bc = t >> 10, pos = t & 1023;
  int pi = pos >> 5, pj = pos & 31;
  float m = (!(pi & 1) && !(pj & 1))
          ? msamp[bc * NPTS + (pi >> 1) * 16 + (pj >> 1)]
          : cam_raw[t];
  m = fminf(fmaxf(m, 0.f), 1.f);
  out_cam[t]  = m;
  out_conf[t] = 1.f / (1.f + expf(-(scale[0] * m + biasp[0])));
}

// ---------------------------------------------------------------------------
// Posterior branch (tiny)
// ---------------------------------------------------------------------------
__global__ __launch_bounds__(256)
void pool_features(const float* __restrict__ feat, float* __restrict__ fg) {
  int wave = threadIdx.x >> 5, lane = threadIdx.x & 31;
  int idx = blockIdx.x * 8 + wave;                 // b*D + d, total B_*D_
  if (idx >= B_ * D_) return;
  const float* p = feat + (size_t)idx * HW_;
  float s = 0.f;
  for (int i = lane; i < HW_; i += 32) s += p[i];
  for (int off = 16; off; off >>= 1) s += __shfl_down(s, off, 32);
  if (!lane) fg[idx] = s * (1.f / (float)HW_);
}

__global__ void mlp_hidden(const float* __restrict__ fg, const float* __restrict__ W1,
                           const float* __restrict__ b1, float* __restrict__ hid) {
  int t = blockIdx.x * blockDim.x + threadIdx.x;   // B_*256
  if (t >= B_ * 256) return;
  int b = t >> 8, o = t & 255;
  float s = b1[o];
  for (int d = 0; d < D_; ++d) s += fg[b * D_ + d] * W1[(size_t)d * 256 + o];
  hid[t] = fmaxf(s, 0.f);
}

__global__ void mlp_out(const float* __restrict__ hid, const float* __restrict__ W2,
                        const float* __restrict__ b2, float* __restrict__ z) {
  int t = blockIdx.x * blockDim.x + threadIdx.x;   // B_*K_
  if (t >= B_ * K_) return;
  int b = t >> 3, k = t & 7;
  float s = b2[k];
  for (int i = 0; i < 256; ++i) s += hid[b * 256 + i] * W2[i * K_ + k];
  z[t] = s;
}

__global__ void z_sample(const float* __restrict__ zmu, const float* __restrict__ zlog,
                         const float* __restrict__ eps, float* __restrict__ zs) {
  int t = threadIdx.x;
  if (t < B_ * K_) zs[t] = zmu[t] + expf(zlog[t]) * eps[t];
}

// ---------------------------------------------------------------------------
// Launcher
// ---------------------------------------------------------------------------
extern "C" void kernel_launch(void* const* d_in, const int* in_sizes, int n_in,
                              void* d_out, int out_size, void* d_ws, size_t ws_size,
                              hipStream_t stream) {
  (void)in_sizes; (void)n_in; (void)out_size; (void)ws_size;
  const float* features  = (const float*)d_in[0];
  const float* cam_raw   = (const float*)d_in[1];
  const float* eps       = (const float*)d_in[3];
  const float* coord_emb = (const float*)d_in[4];
  const float* aff_W1 = (const float*)d_in[5];
  const float* aff_b1 = (const float*)d_in[6];
  const float* aff_W2 = (const float*)d_in[7];
  const float* aff_b2 = (const float*)d_in[8];
  const float* aff_W3 = (const float*)d_in[9];
  const float* aff_b3 = (const float*)d_in[10];
  const float* mu_W1 = (const float*)d_in[11];
  const float* mu_b1 = (const float*)d_in[12];
  const float* mu_W2 = (const float*)d_in[13];
  const float* mu_b2 = (const float*)d_in[14];
  const float* ls_W1 = (const float*)d_in[15];
  const float* ls_b1 = (const float*)d_in[16];
  const float* ls_W2 = (const float*)d_in[17];
  const float* ls_b2 = (const float*)d_in[18];
  const float* conf_scale = (const float*)d_in[19];
  const float* conf_bias  = (const float*)d_in[20];

  float* out = (float*)d_out;
  float* out_cam  = out;                       // (B,C,H,W)      16384
  float* out_zmu  = out + 16384;               // (B,K)             64
  float* out_zlog = out + 16448;               // (B,K)             64
  float* out_zs   = out + 16512;               // (B,K)             64
  float* out_A    = out + 16576;               // (B,C,HW,HW) 16777216
  float* out_conf = out + 16576 + 16777216;    // (B,C,H,W)      16384

  // Workspace carve-out (~76 MB)
  char* w = (char*)d_ws;
  auto take = [&](size_t bytes) { char* r = w; w += (bytes + 255) & ~(size_t)255; return r; };
  int*   ia     = (int*)  take((size_t)SLOTS * 4);
  int*   ib     = (int*)  take((size_t)SLOTS * 4);
  int*   valid  = (int*)  take((size_t)SLOTS * 4);
  float* h1     = (float*)take((size_t)BCN * SLOTS * H1DIM * 4);  // 50.3 MB
  float* h2     = (float*)take((size_t)BCN * SLOTS * H2DIM * 4);  // 25.2 MB
  float* val    = (float*)take((size_t)BCN * SLOTS * 4);
  float* rowsum = (float*)take((size_t)BCN * NPTS * 4);
  float* msamp  = (float*)take((size_t)BCN * NPTS * 4);
  float* fg     = (float*)take((size_t)B_ * D_ * 4);
  float* hidmu  = (float*)take((size_t)B_ * 256 * 4);
  float* hidls  = (float*)take((size_t)B_ * 256 * 4);

  // Pair geometry
  init_pairs<<<(SLOTS + 255) / 256, 256, 0, stream>>>(ia, ib, valid);

  // Posterior branch
  pool_features<<<(B_ * D_) / 8, 256, 0, stream>>>(features, fg);
  mlp_hidden<<<(B_ * 256) / 256, 256, 0, stream>>>(fg, mu_W1, mu_b1, hidmu);
  mlp_out<<<1, 64, 0, stream>>>(hidmu, mu_W2, mu_b2, out_zmu);
  mlp_hidden<<<(B_ * 256) / 256, 256, 0, stream>>>(fg, ls_W1, ls_b1, hidls);
  mlp_out<<<1, 64, 0, stream>>>(hidls, ls_W2, ls_b2, out_zlog);
  z_sample<<<1, 64, 0, stream>>>(out_zmu, out_zlog, eps, out_zs);

  // Affinity MLP (WMMA fp32)
  wmma_gemm<0><<<dim3(SLOTS / BM, H1DIM / BN, BCN), 256, 0, stream>>>(
      nullptr, aff_W1, aff_b1, h1, features, coord_emb, ia, ib, valid,
      SLOTS, H1DIM, IN_DIM);
  wmma_gemm<1><<<dim3(SLOTS / BM, H2DIM / BN, BCN), 256, 0, stream>>>(
      h1, aff_W2, aff_b2, h2, features, coord_emb, ia, ib, valid,
      SLOTS, H2DIM, H1DIM);
  gemm3_val<<<(BCN * SLOTS) / 8, 256, 0, stream>>>(h2, aff_W3, aff_b3, val);

  // A matrix (dense output) + row normalization terms
  row_sums<<<BCN, NPTS, 0, stream>>>(val, valid, rowsum);
  fill_A<<<(BCN * HW_ * (HW_ / 4)) / 256, 256, 0, stream>>>(rowsum, out_A);
  scatter_A<<<(BCN * SLOTS) / 256, 256, 0, stream>>>(val, rowsum, ia, ib, valid, out_A);

  // Sparse diffusion + outputs
  diffusion<<<BCN, NPTS, 0, stream>>>(cam_raw, val, rowsum, valid, msamp);
  finalize_cam<<<(BCN * HW_) / 256, 256, 0, stream>>>(cam_raw, msamp, conf_scale,
                                                      conf_bias, out_cam, out_conf);
}